// TripletLoss_38139309588959
// MI455X (gfx1250) — compile-verified
//
#include <hip/hip_runtime.h>
#include <math.h>

// Problem constants (match reference setup_inputs)
#define B_N   8192
#define D_K   256
#define MARGIN_F 0.3f
#define EPS_F    1e-6f
#define BIGF     3.0e38f
#define LDS_STRIDE 258   // 256 + 2: makes 16-row strided b64 LDS reads bank-conflict-free

typedef __attribute__((ext_vector_type(2))) float v2f;
typedef __attribute__((ext_vector_type(8))) float v8f;

// ---------------------------------------------------------------- init
__global__ void tl_init(float* accum) {
    accum[0] = 0.0f;   // sum of per-anchor losses (valid-weighted)
    accum[1] = 0.0f;   // number of valid anchors
}

// ---------------------------------------------------------------- row sq-norms
__global__ __launch_bounds__(256) void tl_norms(const float* __restrict__ E,
                                                float* __restrict__ sqn) {
    const int wave = threadIdx.x >> 5, lane = threadIdx.x & 31;
    const int row  = blockIdx.x * 8 + wave;
    const float* r = E + (size_t)row * D_K;
    float s = 0.0f;
    #pragma unroll
    for (int c = lane; c < D_K; c += 32) { float v = r[c]; s += v * v; }
    #pragma unroll
    for (int m = 16; m; m >>= 1) s += __shfl_xor(s, m, 32);
    if (lane == 0) sqn[row] = s;
}

// ---------------------------------------------------------------- fused Gram (WMMA f32) + hardest-pos/neg mining
// Block: 256 threads (8 waves). Block owns 32 anchor rows (two 16-row WMMA tiles),
// staged in LDS. Waves split the 512 column tiles round-robin. Distance matrix is
// never materialized: each 16x16 dot tile is consumed immediately into per-row
// running (max,argmax)/(min,argmin) with first-occurrence tie-breaking.
__global__ __launch_bounds__(256) void tl_gram(const float* __restrict__ E,
                                               const int*   __restrict__ labels,
                                               const float* __restrict__ sqn,
                                               int* __restrict__ hp,
                                               int* __restrict__ hn,
                                               int* __restrict__ valid) {
    __shared__ float ldsE[32 * LDS_STRIDE];
    __shared__ int   ldsLab[32];
    __shared__ float ldsSq[32];
    __shared__ float sPV[8][32]; __shared__ int sPI[8][32];
    __shared__ float sNV[8][32]; __shared__ int sNI[8][32];

    const int tid  = threadIdx.x;
    const int wave = tid >> 5, lane = tid & 31;
    const int rowBase = blockIdx.x * 32;

    // Stage this block's 32 anchor rows of E into padded LDS
    for (int idx = tid; idx < 32 * D_K; idx += 256) {
        const int r = idx >> 8, c = idx & 255;
        ldsE[r * LDS_STRIDE + c] = E[(size_t)(rowBase + r) * D_K + c];
    }
    if (tid < 32) { ldsLab[tid] = labels[rowBase + tid]; ldsSq[tid] = sqn[rowBase + tid]; }
    __syncthreads();

    // f32 16x16x4 fragment lane mapping: M/N = lane&15, K-pair base = 2*(lane>>4)
    const int n  = lane & 15;
    const int kb = (lane >> 4) << 1;
    const int halfOfs = (lane >> 4) << 3;   // +8 rows for upper half-wave in C/D layout

    // Running best per lane: 2 row tiles x 8 accumulator rows each
    float pV[2][8]; int pI[2][8]; float nV[2][8]; int nI[2][8];
    #pragma unroll
    for (int t = 0; t < 2; ++t)
        #pragma unroll
        for (int r = 0; r < 8; ++r) { pV[t][r] = -1.0f; pI[t][r] = 0; nV[t][r] = BIGF; nI[t][r] = 0; }

    const float* ldsA0 = ldsE + (size_t)n * LDS_STRIDE + kb;          // rows 0..15
    const float* ldsA1 = ldsA0 + 16 * LDS_STRIDE;                     // rows 16..31

    for (int ct = wave; ct < B_N / 16; ct += 8) {          // uniform trip count per wave
        const int colBase = ct * 16;
        const int j    = colBase + n;                      // this lane's column index
        const int   cLab = labels[j];
        const float cSq  = sqn[j];
        const float* Bp  = E + (size_t)j * D_K + kb;       // B = E^T fragment source

        // Warm L2/WGP$ for the next column tile this wave will touch
        if (ct + 8 < B_N / 16)
            __builtin_prefetch(E + (size_t)((ct + 8) * 16 + n) * D_K, 0, 1);

        v8f acc0 = {}; v8f acc1 = {};
        #pragma unroll 8
        for (int k0 = 0; k0 < D_K; k0 += 4) {
            const v2f b  = *(const v2f*)(Bp    + k0);
            const v2f a0 = *(const v2f*)(ldsA0 + k0);
            const v2f a1 = *(const v2f*)(ldsA1 + k0);
            acc0 = __builtin_amdgcn_wmma_f32_16x16x4_f32(false, a0, false, b, (short)0, acc0, false, false);
            acc1 = __builtin_amdgcn_wmma_f32_16x16x4_f32(false, a1, false, b, (short)0, acc1, false, false);
        }

        // Consume the 16x16 dot tiles: dist^2 = |a|^2 + |b|^2 - 2ab, masked mining
        #pragma unroll
        for (int t = 0; t < 2; ++t) {
            #pragma unroll
            for (int r = 0; r < 8; ++r) {
                const float dot = (t == 0) ? acc0[r] : acc1[r];
                const int   lr  = t * 16 + r + halfOfs;          // local anchor row
                float d = ldsSq[lr] + cSq - 2.0f * dot;
                d = fmaxf(d, 0.0f);
                const bool same = (ldsLab[lr] == cLab);
                const bool diag = (rowBase + lr) == j;
                if (same && !diag) { if (d > pV[t][r]) { pV[t][r] = d; pI[t][r] = j; } }
                if (!same)         { if (d < nV[t][r]) { nV[t][r] = d; nI[t][r] = j; } }
            }
        }
    }

    // Reduce across the 16 lanes sharing each (row, half): xor masks < 16 stay in-half
    #pragma unroll
    for (int t = 0; t < 2; ++t) {
        #pragma unroll
        for (int r = 0; r < 8; ++r) {
            float v = pV[t][r]; int i = pI[t][r];
            #pragma unroll
            for (int m = 1; m < 16; m <<= 1) {
                const float ov = __shfl_xor(v, m, 32); const int oi = __shfl_xor(i, m, 32);
                if (ov > v || (ov == v && oi < i)) { v = ov; i = oi; }
            }
            pV[t][r] = v; pI[t][r] = i;
            v = nV[t][r]; i = nI[t][r];
            #pragma unroll
            for (int m = 1; m < 16; m <<= 1) {
                const float ov = __shfl_xor(v, m, 32); const int oi = __shfl_xor(i, m, 32);
                if (ov < v || (ov == v && oi < i)) { v = ov; i = oi; }
            }
            nV[t][r] = v; nI[t][r] = i;
        }
    }

    if ((lane & 15) == 0) {
        #pragma unroll
        for (int t = 0; t < 2; ++t)
            #pragma unroll
            for (int r = 0; r < 8; ++r) {
                const int lr = t * 16 + r + halfOfs;
                sPV[wave][lr] = pV[t][r]; sPI[wave][lr] = pI[t][r];
                sNV[wave][lr] = nV[t][r]; sNI[wave][lr] = nI[t][r];
            }
    }
    __syncthreads();

    if (tid < 32) {
        float bpv = -1.0f; int bpi = 0; float bnv = BIGF; int bni = 0;
        #pragma unroll
        for (int w = 0; w < 8; ++w) {
            float v = sPV[w][tid]; int i = sPI[w][tid];
            if (v > bpv || (v == bpv && i < bpi)) { bpv = v; bpi = i; }
            v = sNV[w][tid]; i = sNI[w][tid];
            if (v < bnv || (v == bnv && i < bni)) { bnv = v; bni = i; }
        }
        const int gi = rowBase + tid;
        hp[gi]    = bpi;
        hn[gi]    = bni;
        valid[gi] = (bpv >= 0.0f && bnv < BIGF) ? 1 : 0;
    }
}

// ---------------------------------------------------------------- per-anchor loss + reduction
__global__ __launch_bounds__(256) void tl_finalize(const float* __restrict__ E,
                                                   const int* __restrict__ hp,
                                                   const int* __restrict__ hn,
                                                   const int* __restrict__ valid,
                                                   float* accum) {
    const int wave = threadIdx.x >> 5, lane = threadIdx.x & 31;
    const int i = blockIdx.x * 8 + wave;
    const float* a  = E + (size_t)i * D_K;
    const float* p  = E + (size_t)hp[i] * D_K;
    const float* nn = E + (size_t)hn[i] * D_K;
    float sp = 0.0f, sn = 0.0f;
    #pragma unroll
    for (int c = lane; c < D_K; c += 32) {
        const float av = a[c];
        const float dp = av - p[c]  + EPS_F; sp += dp * dp;   // torch pairwise_distance eps
        const float dn = av - nn[c] + EPS_F; sn += dn * dn;
    }
    #pragma unroll
    for (int m = 16; m; m >>= 1) { sp += __shfl_xor(sp, m, 32); sn += __shfl_xor(sn, m, 32); }
    if (lane == 0) {
        const float per = fmaxf(sqrtf(sp) - sqrtf(sn) + MARGIN_F, 0.0f);
        const float v = (float)valid[i];
        atomicAdd(&accum[0], per * v);
        atomicAdd(&accum[1], v);
    }
}

__global__ void tl_writeout(const float* accum, float* out) {
    const float nv = accum[1];
    out[0] = (nv > 0.0f) ? (accum[0] / fmaxf(nv, 1.0f)) : 0.0f;
}

// ---------------------------------------------------------------- host entry
extern "C" void kernel_launch(void* const* d_in, const int* in_sizes, int n_in,
                              void* d_out, int out_size, void* d_ws, size_t ws_size,
                              hipStream_t stream) {
    const float* E      = (const float*)d_in[0];
    const int*   labels = (const int*)d_in[1];
    float* out = (float*)d_out;

    float* sqn   = (float*)d_ws;            // 8192 f
    int*   hp    = (int*)(sqn + B_N);       // 8192 i
    int*   hn    = hp + B_N;                // 8192 i
    int*   valid = hn + B_N;                // 8192 i
    float* accum = (float*)(valid + B_N);   // 2 f

    tl_init    <<<1, 1, 0, stream>>>(accum);
    tl_norms   <<<B_N / 8,  256, 0, stream>>>(E, sqn);
    tl_gram    <<<B_N / 32, 256, 0, stream>>>(E, labels, sqn, hp, hn, valid);
    tl_finalize<<<B_N / 8,  256, 0, stream>>>(E, hp, hn, valid, accum);
    tl_writeout<<<1, 1, 0, stream>>>(accum, out);
}